// AttentionHead_40175124087492
// MI455X (gfx1250) — compile-verified
//
#include <hip/hip_runtime.h>

// Flash attention for B=16, T=2048, d_model=32, d_head=4, causal, fp32.
//
// Roofline: naive scores tensor = 268 MB of HBM traffic (~25+ us at 23.3 TB/s);
// fused flash form touches ~6 MB of HBM, K/V re-reads hit the 192 MB L2.
// d_head==4 maps exactly onto V_WMMA_F32_16X16X4_F32 (A 16x4, B 4x16, C 16x16).
//
// Kernel 1: per-wave 16-row tile computes Q,K,V in ONE wmma accumulator
//           (Wq|Wk|Wv packed into B columns 0-3 / 4-7 / 8-11), 8 wmmas.
// Kernel 2: per-wave (batch, 16-query-tile) flash loop over key tiles:
//           1 wmma for S = Q*K^T, online softmax via shfl_xor half-wave
//           reductions, P transposed through a 1KB per-wave LDS slab,
//           4 wmmas for O += P*V.
//
// All loads are predicated by VALUE (multiply-by-zero mask) instead of EXEC
// branches, so the hot loop has no s_and_saveexec/cbranch overhead and EXEC
// is all-ones at every WMMA (ISA requirement).

typedef float v2f __attribute__((ext_vector_type(2)));
typedef float v8f __attribute__((ext_vector_type(8)));

#define B_N   16
#define T_N   2048
#define DM    32
#define DH    4
#define TILES (T_N / 16)           // 128 query tiles per batch
#define SCALE 0.17677669529663689f // 1/sqrt(32)

// ---------------------------------------------------------------------------
// Kernel 1: fused QKV projection. One wave handles 16 rows of flattened [B*T].
// D[16x16]: cols 0-3 = Q, 4-7 = K, 8-11 = V, 12-15 unused (zero B columns).
// ---------------------------------------------------------------------------
__global__ __launch_bounds__(128) void qkv_proj_kernel(
    const float* __restrict__ q,
    const float* __restrict__ Wq, const float* __restrict__ Wk,
    const float* __restrict__ Wv,
    float* __restrict__ Qo, float* __restrict__ Ko, float* __restrict__ Vo) {
  const int lane = threadIdx.x & 31;
  const int wave = (blockIdx.x * blockDim.x + threadIdx.x) >> 5; // 0..2047
  const int row0 = wave * 16;        // first of 16 rows in flattened [B*T]
  const int m    = lane & 15;        // A row / B,C column index
  const int h    = lane >> 4;        // half-wave
  const int koff = 2 * h;            // K-slice held by this half

  // Branch-free weight selection: lanes 12-15 read Wq but multiply by 0.
  const float* Wsel = (m < 4) ? Wq : (m < 8) ? Wk : (m < 12) ? Wv : Wq;
  const float  wmask = (m < 12) ? 1.0f : 0.0f;
  const int    col   = m & 3;

  v8f acc = {};
#pragma unroll
  for (int c = 0; c < 8; ++c) {             // d_model in chunks of 4
    const int k0 = c * 4 + koff;
    const float* qp = q + (size_t)(row0 + m) * DM + k0;
    v2f a; a.x = qp[0]; a.y = qp[1];        // A[m][k0], A[m][k0+1]
    v2f b;
    b.x = Wsel[k0 * DH + col] * wmask;      // B[koff][m]
    b.y = Wsel[(k0 + 1) * DH + col] * wmask;
    acc = __builtin_amdgcn_wmma_f32_16x16x4_f32(false, a, false, b,
                                                (short)0, acc, false, false);
  }

#pragma unroll
  for (int r = 0; r < 8; ++r) {             // D[r+8h][m]
    const int row = row0 + r + 8 * h;
    const float v = acc[r];
    if (m < 4)       Qo[row * DH + m]        = v;
    else if (m < 8)  Ko[row * DH + (m - 4)]  = v;
    else if (m < 12) Vo[row * DH + (m - 8)]  = v;
  }
}

// ---------------------------------------------------------------------------
// Kernel 2: causal flash attention. One wave per (batch, query tile).
// ---------------------------------------------------------------------------
__global__ __launch_bounds__(128) void flash_attn_kernel(
    const float* __restrict__ Qp, const float* __restrict__ Kp,
    const float* __restrict__ Vp, float* __restrict__ out) {
  __shared__ float ldsP[4][16 * 16];        // 1KB per wave: P transpose slab
  const int lane = threadIdx.x & 31;
  const int wloc = threadIdx.x >> 5;
  const int wave = blockIdx.x * 4 + wloc;   // 0..2047
  const int b    = wave / TILES;
  const int it   = wave % TILES;            // query tile index
  const int t0   = it * 16;
  const int m    = lane & 15;
  const int h    = lane >> 4;
  const int koff = 2 * h;
  float* P = ldsP[wloc];

  // Q A-operand for this tile, softmax scale folded in.
  v2f aq;
  {
    const float* qp = Qp + (size_t)(b * T_N + t0 + m) * DH + koff;
    aq.x = qp[0] * SCALE; aq.y = qp[1] * SCALE;
  }

  // Branch-free V-operand masking: lanes m>=4 read col (m&3) but scale by 0.
  const float vmask = (m < 4) ? 1.0f : 0.0f;
  const int   vcol  = m & 3;

  float mi[8], li[8];
#pragma unroll
  for (int r = 0; r < 8; ++r) { mi[r] = -1e30f; li[r] = 0.f; }
  v8f o = {};                               // O accumulator, cols 0-3 valid

  for (int j = 0; j <= it; ++j) {           // causal: key tiles 0..it
    // K tile as B operand: B[k][n] = K[j*16+n][k]
    v2f bk;
    {
      const float* kp = Kp + (size_t)(b * T_N + j * 16 + m) * DH + koff;
      bk.x = kp[0]; bk.y = kp[1];
    }
    v8f zc = {};
    v8f s = __builtin_amdgcn_wmma_f32_16x16x4_f32(false, aq, false, bk,
                                                  (short)0, zc, false, false);

    // Causal mask only on the diagonal tile: key n <= query r+8h.
    if (j == it) {
#pragma unroll
      for (int r = 0; r < 8; ++r)
        if (m > r + 8 * h) s[r] = -1e30f;
    }

    // Online softmax: per-row (r+8h) stats via 16-lane half-wave reductions.
#pragma unroll
    for (int r = 0; r < 8; ++r) {
      float v = s[r];
      v = fmaxf(v, __shfl_xor(v, 1, 32));
      v = fmaxf(v, __shfl_xor(v, 2, 32));
      v = fmaxf(v, __shfl_xor(v, 4, 32));
      v = fmaxf(v, __shfl_xor(v, 8, 32));
      const float mnew  = fmaxf(mi[r], v);
      const float p     = __expf(s[r] - mnew);
      float ps = p;
      ps += __shfl_xor(ps, 1, 32);
      ps += __shfl_xor(ps, 2, 32);
      ps += __shfl_xor(ps, 4, 32);
      ps += __shfl_xor(ps, 8, 32);
      const float alpha = __expf(mi[r] - mnew);
      li[r] = li[r] * alpha + ps;
      mi[r] = mnew;
      o[r] *= alpha;
      P[(r + 8 * h) * 16 + m] = p;          // C-layout -> LDS (row*16+col)
    }

    // O += P * V as 4 wmmas over key chunks of 4 (dscnt-ordered LDS reload).
#pragma unroll
    for (int c = 0; c < 4; ++c) {
      const float* pp = &P[m * 16 + c * 4 + koff]; // A[m][4c+koff..]
      v2f ap; ap.x = pp[0]; ap.y = pp[1];
      const float* vp =
          Vp + (size_t)(b * T_N + j * 16 + c * 4 + koff) * DH + vcol;
      v2f bv;                                      // B[k][n]=V[j*16+4c+k][n]
      bv.x = vp[0]  * vmask;
      bv.y = vp[DH] * vmask;
      o = __builtin_amdgcn_wmma_f32_16x16x4_f32(false, ap, false, bv,
                                                (short)0, o, false, false);
    }
  }

  // Normalize and store: D[r+8h][m], m<4 are the valid head dims.
#pragma unroll
  for (int r = 0; r < 8; ++r) {
    if (m < 4)
      out[(size_t)(b * T_N + t0 + r + 8 * h) * DH + m] = o[r] / li[r];
  }
}

// ---------------------------------------------------------------------------
extern "C" void kernel_launch(void* const* d_in, const int* in_sizes, int n_in,
                              void* d_out, int out_size, void* d_ws,
                              size_t ws_size, hipStream_t stream) {
  (void)in_sizes; (void)n_in; (void)out_size; (void)ws_size;
  const float* q  = (const float*)d_in[0];
  const float* Wq = (const float*)d_in[1];
  const float* Wk = (const float*)d_in[2];
  const float* Wv = (const float*)d_in[3];
  float* ws = (float*)d_ws;                 // 3 * B*T*DH floats = 1.5 MB
  float* Qo = ws;
  float* Ko = Qo + (size_t)B_N * T_N * DH;
  float* Vo = Ko + (size_t)B_N * T_N * DH;
  float* out = (float*)d_out;

  const int waves  = B_N * TILES;           // 2048 waves total
  const int blocks = waves / 4;             // 4 waves (128 threads) per block

  qkv_proj_kernel<<<blocks, 128, 0, stream>>>(q, Wq, Wk, Wv, Qo, Ko, Vo);
  flash_attn_kernel<<<blocks, 128, 0, stream>>>(Qo, Ko, Vo, out);
}